// DMoNHead_66632122630722
// MI455X (gfx1250) — compile-verified
//
#include <hip/hip_runtime.h>
#include <hip/hip_bf16.h>

typedef __attribute__((ext_vector_type(2))) float v2f;
typedef __attribute__((ext_vector_type(8))) float v8f;

#define N_PTS 8192
#define D_DIM 256
#define H_DIM 256
#define K_CL  64
#define NTYPES 17

// ---- workspace layout (in 32-bit words) ----
// [0, 2097152)            : adjacency bitmask (8192 rows x 256 words)
// [2097152, 2105344)      : deg (float per node)
// [2105344, ...)          : accumulators:
//    +0   edgeSum, +1 degSum, +2 cluster_size[64], +66 st_d[64],
//    +130 type_cluster[17*64], +1218 ss[64*64]
#define WS_ADJ_WORDS   (N_PTS * (N_PTS / 32))
#define WS_DEG_OFF     WS_ADJ_WORDS
#define WS_ACC_OFF     (WS_ADJ_WORDS + N_PTS)
#define ACC_WORDS      (2 + 64 + 64 + NTYPES * 64 + 64 * 64)
#define WS_TOTAL_WORDS (WS_ACC_OFF + ACC_WORDS)

__global__ void k_zero(unsigned* __restrict__ p, int nWords) {
    int t = blockIdx.x * blockDim.x + threadIdx.x;
    if (t < nWords) p[t] = 0u;
}

// Build symmetric 0/1 adjacency as a bit matrix. atomicOr dedups multi-edges
// and makes min(adj + adj^T, 1) exact (incl. self loops -> single diagonal 1).
__global__ void k_edges(const int* __restrict__ e0, const int* __restrict__ e1,
                        unsigned* __restrict__ adjBits, int E_) {
    int t = blockIdx.x * blockDim.x + threadIdx.x;
    if (t >= E_) return;
    int i = e0[t], j = e1[t];
    atomicOr(adjBits + (size_t)i * (N_PTS / 32) + (j >> 5), 1u << (j & 31));
    atomicOr(adjBits + (size_t)j * (N_PTS / 32) + (i >> 5), 1u << (i & 31));
}

__global__ void k_deg(const unsigned* __restrict__ adjBits,
                      float* __restrict__ deg, float* __restrict__ degSum) {
    int i = blockIdx.x * blockDim.x + threadIdx.x;
    if (i >= N_PTS) return;
    const unsigned* row = adjBits + (size_t)i * (N_PTS / 32);
    int c = 0;
    for (int w = 0; w < N_PTS / 32; ++w) c += __popc(row[w]);
    deg[i] = (float)c;
    atomicAdd(degSum, (float)c);
}

// Fused: h = selu(emb @ W^T + b) -> logits = h @ centers^T * H^-0.5 -> softmax.
// One block per 16 rows, 8 waves. f32 WMMA 16x16x4 keeps exact f32 math.
__global__ __launch_bounds__(256) void k_gemm(
    const float* __restrict__ emb, const float* __restrict__ W,
    const float* __restrict__ bvec, const float* __restrict__ centers,
    float* __restrict__ logits_out, float* __restrict__ s_out) {
    __shared__ float lds_h[16 * H_DIM];      // 16 KB
    __shared__ float lds_logit[16 * K_CL];   // 4 KB
    const int tid  = threadIdx.x;
    const int wave = tid >> 5;
    const int lane = tid & 31;
    const int half = lane >> 4;   // K sub-select per ISA f32 A/B layout
    const int lr   = lane & 15;
    const int r0   = blockIdx.x * 16;

    // --- stage 1: two 16x16 column tiles of h per wave (cols w*16, w*16+128)
    const int c0 = wave * 16;
    const int c1 = c0 + 128;
    v8f acc0 = {}, acc1 = {};
    const float* arow  = emb + (size_t)(r0 + lr) * D_DIM;
    const float* w0row = W + (size_t)(c0 + lr) * D_DIM;
    const float* w1row = W + (size_t)(c1 + lr) * D_DIM;
    for (int k = 0; k < D_DIM; k += 4) {
        v2f a  = *(const v2f*)(arow  + k + 2 * half);
        v2f b0 = *(const v2f*)(w0row + k + 2 * half);
        v2f b1 = *(const v2f*)(w1row + k + 2 * half);
        acc0 = __builtin_amdgcn_wmma_f32_16x16x4_f32(false, a, false, b0, (short)0, acc0, false, false);
        acc1 = __builtin_amdgcn_wmma_f32_16x16x4_f32(false, a, false, b1, (short)0, acc1, false, false);
    }
    const float kSeluScale = 1.0507009873554805f;
    const float kSeluAlpha = 1.6732632423543772f;
    float bb0 = bvec[c0 + lr];
    float bb1 = bvec[c1 + lr];
#pragma unroll
    for (int v = 0; v < 8; ++v) {
        int m = v + 8 * half;            // C/D layout: VGPR v -> M=v (+8 hi half)
        float x0 = acc0[v] + bb0;
        float x1 = acc1[v] + bb1;
        x0 = (x0 > 0.f) ? kSeluScale * x0 : kSeluScale * kSeluAlpha * (__expf(x0) - 1.f);
        x1 = (x1 > 0.f) ? kSeluScale * x1 : kSeluScale * kSeluAlpha * (__expf(x1) - 1.f);
        lds_h[m * H_DIM + c0 + lr] = x0;
        lds_h[m * H_DIM + c1 + lr] = x1;
    }
    __syncthreads();

    // --- stage 2: logits (16 x 64), 4 tiles on waves 0..3
    if (wave < 4) {
        v8f accL = {};
        const float* crow = centers + (size_t)(wave * 16 + lr) * H_DIM;
        for (int k = 0; k < H_DIM; k += 4) {
            v2f a;
            a.x = lds_h[lr * H_DIM + k + 2 * half];
            a.y = lds_h[lr * H_DIM + k + 2 * half + 1];
            v2f b = *(const v2f*)(crow + k + 2 * half);
            accL = __builtin_amdgcn_wmma_f32_16x16x4_f32(false, a, false, b, (short)0, accL, false, false);
        }
        const float scl = 0.0625f;  // H^-0.5 = 1/16
#pragma unroll
        for (int v = 0; v < 8; ++v) {
            int m = v + 8 * half;
            lds_logit[m * K_CL + wave * 16 + lr] = accL[v] * scl;
        }
    }
    __syncthreads();

    // --- stage 3: per-row softmax, emit logits and s
    if (tid < 16) {
        const float* lrow = lds_logit + tid * K_CL;
        float mx = lrow[0];
        for (int a2 = 1; a2 < K_CL; ++a2) mx = fmaxf(mx, lrow[a2]);
        float sum = 0.f;
        for (int a2 = 0; a2 < K_CL; ++a2) sum += __expf(lrow[a2] - mx);
        float inv = 1.f / sum;
        size_t base = (size_t)(r0 + tid) * K_CL;
        for (int a2 = 0; a2 < K_CL; ++a2) {
            float lv = lrow[a2];
            logits_out[base + a2] = lv;
            s_out[base + a2] = __expf(lv - mx) * inv;
        }
    }
}

// ss = s^T @ s (64x64, K = 8192) via f32 WMMA; one block, 16 waves = 16 tiles.
__global__ __launch_bounds__(512) void k_ss(const float* __restrict__ s,
                                            float* __restrict__ ss) {
    const int tid  = threadIdx.x;
    const int wave = tid >> 5;
    const int lane = tid & 31;
    const int half = lane >> 4, lr = lane & 15;
    const int mi = (wave >> 2) * 16;
    const int ni = (wave & 3) * 16;
    v8f acc = {};
    for (int k = 0; k < N_PTS; k += 4) {
        v2f a, b;
        size_t k0 = (size_t)(k + 2 * half) * K_CL;
        size_t k1 = (size_t)(k + 2 * half + 1) * K_CL;
        a.x = s[k0 + mi + lr]; a.y = s[k1 + mi + lr];   // A[a,k'] = s[k', mi+a]
        b.x = s[k0 + ni + lr]; b.y = s[k1 + ni + lr];   // B[k',b] = s[k', ni+b]
        acc = __builtin_amdgcn_wmma_f32_16x16x4_f32(false, a, false, b, (short)0, acc, false, false);
    }
#pragma unroll
    for (int v = 0; v < 8; ++v)
        ss[(size_t)(mi + v + 8 * half) * K_CL + ni + lr] = acc[v];
}

// cluster_size, st_d = s^T deg, type_cluster via LDS-staged atomics.
__global__ __launch_bounds__(256) void k_rowstats(
    const float* __restrict__ s, const float* __restrict__ deg,
    const int* __restrict__ jt, float* __restrict__ cs,
    float* __restrict__ sd, float* __restrict__ tc) {
    __shared__ float l_cs[4][64];
    __shared__ float l_sd[4][64];
    __shared__ float l_tc[NTYPES * 64];
    int t = threadIdx.x;
    int a = t & 63, q = t >> 6;
    for (int x = t; x < NTYPES * 64; x += 256) l_tc[x] = 0.f;
    __syncthreads();
    int rbase = blockIdx.x * 64 + q * 16;
    float accC = 0.f, accS = 0.f;
    for (int r = 0; r < 16; ++r) {
        int row = rbase + r;
        float v = s[(size_t)row * K_CL + a];
        accC += v;
        accS += deg[row] * v;
        atomicAdd(&l_tc[jt[row] * 64 + a], v);
    }
    l_cs[q][a] = accC;
    l_sd[q][a] = accS;
    __syncthreads();
    if (q == 0) {
        atomicAdd(&cs[a], l_cs[0][a] + l_cs[1][a] + l_cs[2][a] + l_cs[3][a]);
        atomicAdd(&sd[a], l_sd[0][a] + l_sd[1][a] + l_sd[2][a] + l_sd[3][a]);
    }
    for (int x = t; x < NTYPES * 64; x += 256)
        if (l_tc[x] != 0.f) atomicAdd(&tc[x], l_tc[x]);
}

// trace(s^T A s) = sum over set bits (i,j) of dot(s_i, s_j).
__global__ __launch_bounds__(256) void k_edgesum(
    const unsigned* __restrict__ adjBits, const float* __restrict__ s,
    float* __restrict__ edgeSum) {
    __shared__ float red[256];
    int t = blockIdx.x * blockDim.x + threadIdx.x;  // one thread per bit-word
    int i = t >> 8;
    int w = t & 255;
    unsigned bits = adjBits[t];
    float acc = 0.f;
    const float* si = s + (size_t)i * K_CL;
    while (bits) {
        int b = __builtin_ctz(bits);
        bits &= bits - 1;
        const float* sj = s + (size_t)(w * 32 + b) * K_CL;
        float d = 0.f;
        for (int a2 = 0; a2 < K_CL; ++a2) d += si[a2] * sj[a2];
        acc += d;
    }
    red[threadIdx.x] = acc;
    __syncthreads();
    for (int off = 128; off > 0; off >>= 1) {
        if (threadIdx.x < off) red[threadIdx.x] += red[threadIdx.x + off];
        __syncthreads();
    }
    if (threadIdx.x == 0 && red[0] != 0.f) atomicAdd(edgeSum, red[0]);
}

__global__ __launch_bounds__(256) void k_final(const float* __restrict__ acc,
                                               float* __restrict__ outScalars) {
    __shared__ float red[256];
    int t = threadIdx.x;
    const float* cs = acc + 2;
    const float* sd = acc + 66;
    const float* tc = acc + 130;
    const float* ss = acc + 1218;

    float x = 0.f;                                  // ||ss||_F^2
    for (int idx = t; idx < 4096; idx += 256) x += ss[idx] * ss[idx];
    red[t] = x; __syncthreads();
    for (int off = 128; off > 0; off >>= 1) { if (t < off) red[t] += red[t + off]; __syncthreads(); }
    float ssn = sqrtf(red[0]) + 1e-8f;
    __syncthreads();

    x = 0.f;                                        // ||ss/||ss|| - I/sqrt(k)||_F
    for (int idx = t; idx < 4096; idx += 256) {
        int r = idx >> 6, c = idx & 63;
        float v = ss[idx] / ssn - ((r == c) ? 0.125f : 0.f);
        x += v * v;
    }
    red[t] = x; __syncthreads();
    for (int off = 128; off > 0; off >>= 1) { if (t < off) red[t] += red[t + off]; __syncthreads(); }
    float ortho = sqrtf(red[0]);
    __syncthreads();

    x = (t < 64) ? cs[t] * cs[t] : 0.f;             // ||cluster_size||^2
    red[t] = x; __syncthreads();
    for (int off = 128; off > 0; off >>= 1) { if (t < off) red[t] += red[t + off]; __syncthreads(); }
    float clusterLoss = (8.f / 8192.f) * sqrtf(red[0]) - 1.f;
    __syncthreads();

    x = (t < 64) ? sd[t] * sd[t] : 0.f;             // ||s^T deg||^2
    red[t] = x; __syncthreads();
    for (int off = 128; off > 0; off >>= 1) { if (t < off) red[t] += red[t + off]; __syncthreads(); }
    float sdn2 = red[0];
    __syncthreads();

    x = 0.f;                                        // type loss
    for (int idx = t; idx < NTYPES * 64; idx += 256) {
        float v = tc[idx] - 1.f;
        if (v > 0.f) x += v * v;
    }
    red[t] = x; __syncthreads();
    for (int off = 128; off > 0; off >>= 1) { if (t < off) red[t] += red[t + off]; __syncthreads(); }
    float typeLoss = red[0];

    if (t == 0) {
        float m = acc[1] * 0.5f;
        float spectral = (m < 1e-8f) ? 0.f : -(acc[0] - sdn2 / (2.f * m)) / (2.f * m);
        outScalars[0] = spectral;
        outScalars[1] = ortho;
        outScalars[2] = clusterLoss;
        outScalars[3] = typeLoss;
    }
}

extern "C" void kernel_launch(void* const* d_in, const int* in_sizes, int n_in,
                              void* d_out, int out_size, void* d_ws, size_t ws_size,
                              hipStream_t stream) {
    const float* emb = (const float*)d_in[0];
    const float* W   = (const float*)d_in[1];
    const float* bv  = (const float*)d_in[2];
    const float* cp  = (const float*)d_in[3];
    const int* eidx  = (const int*)d_in[4];
    const int* jt    = (const int*)d_in[5];
    const int E_ = in_sizes[4] / 2;

    unsigned* adjBits = (unsigned*)d_ws;
    float* deg = (float*)d_ws + WS_DEG_OFF;
    float* acc = (float*)d_ws + WS_ACC_OFF;
    float* edgeSum = acc + 0;
    float* degSum  = acc + 1;
    float* cs      = acc + 2;
    float* sd      = acc + 66;
    float* tc      = acc + 130;
    float* ss      = acc + 1218;

    float* logits_out = (float*)d_out;
    float* s_out      = logits_out + (size_t)N_PTS * K_CL;
    float* scalars    = s_out + (size_t)N_PTS * K_CL;

    k_zero<<<(WS_TOTAL_WORDS + 255) / 256, 256, 0, stream>>>((unsigned*)d_ws, WS_TOTAL_WORDS);
    k_edges<<<(E_ + 255) / 256, 256, 0, stream>>>(eidx, eidx + E_, adjBits, E_);
    k_deg<<<(N_PTS + 255) / 256, 256, 0, stream>>>(adjBits, deg, degSum);
    k_gemm<<<N_PTS / 16, 256, 0, stream>>>(emb, W, bv, cp, logits_out, s_out);
    k_ss<<<1, 512, 0, stream>>>(s_out, ss);
    k_rowstats<<<N_PTS / 64, 256, 0, stream>>>(s_out, deg, jt, cs, sd, tc);
    k_edgesum<<<(N_PTS * (N_PTS / 32)) / 256, 256, 0, stream>>>(adjBits, s_out, edgeSum);
    k_final<<<1, 256, 0, stream>>>(acc, scalars);
}